// RealInvo_2765958938662
// MI455X (gfx1250) — compile-verified
//
#include <hip/hip_runtime.h>
#include <hip/hip_bf16.h>

typedef __attribute__((ext_vector_type(2))) float v2f;
typedef __attribute__((ext_vector_type(8))) float v8f;

#define Bn 4
#define Cc 256
#define Hh 56
#define Ww 56
#define HW (Hh * Ww)          // 3136
#define NP (Bn * HW)          // 12544 total pixels
#define TN 256                // TNUM
#define Gg 16
#define Kk 7
#define KK (Kk * Kk)          // 49
#define NK (Gg * KK)          // 784
#define EPSf 1e-5f

// ---------------------------------------------------------------------------
// Kernel 1: t[p][n] = PReLU(BN(sum_k x[p][k] * reduce_w[n][k] + rb[n]))
// 2x2 register blocking: one wave computes a 32(M=pixels) x 32(N=channels)
// output block = 4 accumulator tiles, 4 WMMAs per K-step of 4, sharing the
// A and B fragments pairwise (halves load traffic per WMMA).
// f32 WMMA layouts (CDNA5 ISA):
//   A 16x4 : lane L -> M=L%16, vgpr v -> K=2*(L/16)+v
//   B 4x16 : lane L -> N=L%16, vgpr v -> K=2*(L/16)+v
//   D 16x16: vgpr r, lane L -> M=r+8*(L/16), N=L%16
// ---------------------------------------------------------------------------
__global__ __launch_bounds__(256) void gemm1_reduce_bn_prelu(
    const float* __restrict__ x, const float* __restrict__ rw,
    const float* __restrict__ rb, const float* __restrict__ gamma,
    const float* __restrict__ beta, const float* __restrict__ mean,
    const float* __restrict__ var, const float* __restrict__ pa,
    float* __restrict__ t_out)
{
    const int lane  = threadIdx.x & 31;
    const int wave  = threadIdx.x >> 5;
    const int tile  = blockIdx.x * 8 + wave;     // 392*8 = 3136 pair-tiles
    const int npair = tile & 7;                  // 8 N-pairs (32 chans each)
    const int mpair = tile >> 3;                 // 392 M-pairs (32 pix each)
    const int half  = lane >> 4;
    const int l16   = lane & 15;

    const int p_base = mpair * 32;               // 3136%32==0 -> no batch cross
    const int b      = p_base / HW;
    const int hw     = p_base - b * HW;
    const float* xrow0 = x + (size_t)b * Cc * HW + hw + l16;   // M tile 0
    const float* xrow1 = xrow0 + 16;                           // M tile 1
    const int n0 = npair * 32 + l16;                           // N tile 0
    const int n1 = n0 + 16;                                    // N tile 1
    const float* wrow0 = rw + (size_t)n0 * Cc;
    const float* wrow1 = rw + (size_t)n1 * Cc;

    v8f acc00 = {}, acc01 = {}, acc10 = {}, acc11 = {};
#pragma unroll 4
    for (int k0 = 0; k0 < Cc; k0 += 4) {
        const int kh = k0 + 2 * half;
        v2f a0, a1;
        a0.x = xrow0[(size_t)kh * HW];
        a0.y = xrow0[(size_t)(kh + 1) * HW];
        a1.x = xrow1[(size_t)kh * HW];
        a1.y = xrow1[(size_t)(kh + 1) * HW];
        v2f b0 = *(const v2f*)(wrow0 + kh);      // contiguous, 8B-aligned
        v2f b1 = *(const v2f*)(wrow1 + kh);
        acc00 = __builtin_amdgcn_wmma_f32_16x16x4_f32(false, a0, false, b0, (short)0, acc00, false, false);
        acc01 = __builtin_amdgcn_wmma_f32_16x16x4_f32(false, a0, false, b1, (short)0, acc01, false, false);
        acc10 = __builtin_amdgcn_wmma_f32_16x16x4_f32(false, a1, false, b0, (short)0, acc10, false, false);
        acc11 = __builtin_amdgcn_wmma_f32_16x16x4_f32(false, a1, false, b1, (short)0, acc11, false, false);
    }

    // epilogue: bias + BN(eval) + PReLU, params depend only on N (per lane)
    const float al = pa[0];
    const float s0  = gamma[n0] * rsqrtf(var[n0] + EPSf);
    const float s1  = gamma[n1] * rsqrtf(var[n1] + EPSf);
    const float bi0 = rb[n0] - mean[n0], bi1 = rb[n1] - mean[n1];
    const float bt0 = beta[n0],          bt1 = beta[n1];

    float* trow0 = t_out + (size_t)(p_base + 8 * half) * TN;        // M tile 0
    float* trow1 = t_out + (size_t)(p_base + 16 + 8 * half) * TN;   // M tile 1
#pragma unroll
    for (int r = 0; r < 8; ++r) {
        float v00 = (acc00[r] + bi0) * s0 + bt0; v00 = v00 > 0.f ? v00 : al * v00;
        float v01 = (acc01[r] + bi1) * s1 + bt1; v01 = v01 > 0.f ? v01 : al * v01;
        float v10 = (acc10[r] + bi0) * s0 + bt0; v10 = v10 > 0.f ? v10 : al * v10;
        float v11 = (acc11[r] + bi1) * s1 + bt1; v11 = v11 > 0.f ? v11 : al * v11;
        trow0[(size_t)r * TN + n0] = v00;
        trow0[(size_t)r * TN + n1] = v01;
        trow1[(size_t)r * TN + n0] = v10;
        trow1[(size_t)r * TN + n1] = v11;
    }
}

// ---------------------------------------------------------------------------
// Kernel 2: ker[kk][p] = sum_k span_w[kk][k] * t[p][k] + sb[kk]
// Swapped orientation (M = kk rows, N = pixels) so both operands are aligned
// float2 loads, with 1x4 N-blocking: one A (span_w) fragment is reused across
// 4 pixel tiles -> 4 WMMAs per 5 loads, and span_w refetch traffic /4.
// Output stored [kk][NP] so the involution kernel reads it coalesced.
// ---------------------------------------------------------------------------
__global__ __launch_bounds__(128) void gemm2_span(
    const float* __restrict__ t_in, const float* __restrict__ sw,
    const float* __restrict__ sb, float* __restrict__ ker)
{
    const int lane  = threadIdx.x & 31;
    const int wave  = threadIdx.x >> 5;
    const int tile  = blockIdx.x * 4 + wave;     // 49*196 = 9604 tiles
    const int mtile = tile % 49;                 // kk tiles
    const int nquad = tile / 49;                 // groups of 4 pixel tiles
    const int half  = lane >> 4;
    const int l16   = lane & 15;

    const float* arow = sw + (size_t)(mtile * 16 + l16) * TN;   // A: M=l16
    const int p_base  = nquad * 64;
    const float* brow0 = t_in + (size_t)(p_base +  0 + l16) * TN;
    const float* brow1 = t_in + (size_t)(p_base + 16 + l16) * TN;
    const float* brow2 = t_in + (size_t)(p_base + 32 + l16) * TN;
    const float* brow3 = t_in + (size_t)(p_base + 48 + l16) * TN;

    v8f acc0 = {}, acc1 = {}, acc2 = {}, acc3 = {};
#pragma unroll 4
    for (int k0 = 0; k0 < TN; k0 += 4) {
        const int kh = k0 + 2 * half;
        v2f a  = *(const v2f*)(arow + kh);
        v2f b0 = *(const v2f*)(brow0 + kh);
        v2f b1 = *(const v2f*)(brow1 + kh);
        v2f b2 = *(const v2f*)(brow2 + kh);
        v2f b3 = *(const v2f*)(brow3 + kh);
        acc0 = __builtin_amdgcn_wmma_f32_16x16x4_f32(false, a, false, b0, (short)0, acc0, false, false);
        acc1 = __builtin_amdgcn_wmma_f32_16x16x4_f32(false, a, false, b1, (short)0, acc1, false, false);
        acc2 = __builtin_amdgcn_wmma_f32_16x16x4_f32(false, a, false, b2, (short)0, acc2, false, false);
        acc3 = __builtin_amdgcn_wmma_f32_16x16x4_f32(false, a, false, b3, (short)0, acc3, false, false);
    }

    const int kkbase = mtile * 16 + 8 * half;
#pragma unroll
    for (int r = 0; r < 8; ++r) {
        const int kkr = kkbase + r;
        const float bias = sb[kkr];
        float* krow = ker + (size_t)kkr * NP + p_base + l16;
        krow[ 0] = acc0[r] + bias;
        krow[16] = acc1[r] + bias;
        krow[32] = acc2[r] + bias;
        krow[48] = acc3[r] + bias;
    }
}

// ---------------------------------------------------------------------------
// Kernel 3: involution. One thread per (b, g, pixel); the 49 ker values are
// loaded once into registers (coalesced: consecutive lanes = consecutive p)
// and reused across the 16 channels of the group (x16 traffic reduction).
// x patch loads are coalesced along w; zero padding via bounds check.
// ---------------------------------------------------------------------------
__global__ __launch_bounds__(256) void involution_apply(
    const float* __restrict__ x, const float* __restrict__ ker,
    float* __restrict__ out)
{
    const int id = blockIdx.x * 256 + threadIdx.x;   // B*G*HW = 200704
    const int hw = id % HW;
    const int bg = id / HW;
    const int g  = bg % Gg;
    const int b  = bg / Gg;
    const int h  = hw / Ww;
    const int w  = hw % Ww;
    const int p  = b * HW + hw;

    float kv[KK];
#pragma unroll
    for (int idx = 0; idx < KK; ++idx)
        kv[idx] = ker[(size_t)(g * KK + idx) * NP + p];

    const float* xb = x   + (size_t)b * Cc * HW;
    float*       ob = out + (size_t)b * Cc * HW;
    for (int cc = 0; cc < Gg; ++cc) {
        const int c = g * 16 + cc;
        const float* xc = xb + (size_t)c * HW;
        float acc = 0.f;
#pragma unroll
        for (int idx = 0; idx < KK; ++idx) {
            const int hh = h + idx / Kk - 3;
            const int ww = w + idx % Kk - 3;
            if (hh >= 0 && hh < Hh && ww >= 0 && ww < Ww)
                acc += xc[hh * Ww + ww] * kv[idx];
        }
        ob[(size_t)c * HW + hw] = acc;
    }
}

extern "C" void kernel_launch(void* const* d_in, const int* in_sizes, int n_in,
                              void* d_out, int out_size, void* d_ws, size_t ws_size,
                              hipStream_t stream) {
    const float* x     = (const float*)d_in[0];
    const float* rw    = (const float*)d_in[1];
    const float* rb    = (const float*)d_in[2];
    const float* gamma = (const float*)d_in[3];
    const float* beta  = (const float*)d_in[4];
    const float* mean  = (const float*)d_in[5];
    const float* var   = (const float*)d_in[6];
    const float* pa    = (const float*)d_in[7];
    const float* sw    = (const float*)d_in[8];
    const float* sb    = (const float*)d_in[9];
    float* out = (float*)d_out;

    float* t_ws   = (float*)d_ws;                      // [NP][TN]  = 12.8 MB
    float* ker_ws = t_ws + (size_t)NP * TN;            // [NK][NP]  = 39.3 MB

    // GEMM1: 392 M-pairs * 8 N-pairs = 3136 waves -> 392 blocks of 8 waves
    gemm1_reduce_bn_prelu<<<392, 256, 0, stream>>>(
        x, rw, rb, gamma, beta, mean, var, pa, t_ws);
    // GEMM2: 49 * 196 = 9604 waves -> 2401 blocks of 4 waves (128 threads)
    gemm2_span<<<2401, 128, 0, stream>>>(
        t_ws, sw, sb, ker_ws);
    // Involution: 200704 threads -> 784 blocks (exact)
    involution_apply<<<(Bn * Gg * HW) / 256, 256, 0, stream>>>(
        x, ker_ws, out);
}